// TripletLoss_7834020348595
// MI455X (gfx1250) — compile-verified
//
#include <hip/hip_runtime.h>

typedef float v2f __attribute__((ext_vector_type(2)));
typedef float v8f __attribute__((ext_vector_type(8)));

#define B_NETS 64
#define G_GENES 1000
#define D_EMB 8
#define MARGIN_F 1.0f
#define GENE_SPLIT 25
#define GENES_PER_CHUNK (G_GENES / GENE_SPLIT)   // 40 genes per wave

// ---------------------------------------------------------------------------
// Kernel 0: zero the distance accumulators in workspace (2 x 64 x 64 floats)
// ---------------------------------------------------------------------------
__global__ void zero_ws_kernel(float* __restrict__ ws, int n) {
    int i = blockIdx.x * blockDim.x + threadIdx.x;
    if (i < n) ws[i] = 0.0f;
}

// ---------------------------------------------------------------------------
// Kernel 1: per-gene pairwise distances via V_WMMA_F32_16X16X4_F32 with the
// augmented-vector trick:
//   A'[m] = [-2*a_0 .. -2*a_7, ||a_m||^2, 1, 0, 0]   (K = 12)
//   B'[n] = [  e_0 ..   e_7,   1, ||e_n||^2, 0, 0]
//   A'.B' = ||a||^2 + ||e||^2 - 2 a.e = dist^2  -> accumulator IS dist^2
// Three chained K=4 WMMA steps; fragment layout (f32 16x4): lane-half h,
// VGPR v holds K = 4s + 2h + v. Third step: half0 = (na,1)/(1,nb), half1 = 0.
//
// grid = (16 tiles [ti*4+tj], 2 matrices [pos/neg], GENE_SPLIT chunks)
// block = 32 (one wave32; WMMA requires EXEC all ones, uniform control flow)
// C/D layout: VGPR r holds (M = r + 8*(lane/16), N = lane%16).
// sqrt uses the raw v_sqrt_f32 (TRANS, co-issues with VALU) instead of the
// IEEE denorm-correct libm expansion (~10 VALU ops each).
// ---------------------------------------------------------------------------
__global__ __launch_bounds__(32) void tile_dist_kernel(
    const float* __restrict__ anchor,
    const float* __restrict__ positive,
    const float* __restrict__ negative,
    float* __restrict__ dist_ws) {
    const int tile = blockIdx.x;          // 0..15
    const int ti   = tile >> 2;
    const int tj   = tile & 3;
    const int mat  = blockIdx.y;          // 0 = pos, 1 = neg
    const int lane = threadIdx.x;         // 0..31
    const int half = lane >> 4;           // 0 or 1
    const int lmod = lane & 15;

    const float* other = (mat == 0) ? positive : negative;
    float* out = dist_ws + mat * (B_NETS * B_NETS);

    const int arow  = ti * 16 + lmod;     // anchor row for A fragment
    const int brow  = tj * 16 + lmod;     // other row for B fragment
    const int kbase = half * 2;           // K offset per lane-half

    const float one_if_h0 = (half == 0) ? 1.0f : 0.0f;  // loop-invariant

    float acc[8];
#pragma unroll
    for (int r = 0; r < 8; ++r) acc[r] = 0.0f;

    const int g0 = blockIdx.z * GENES_PER_CHUNK;
#pragma unroll 2
    for (int g = g0; g < g0 + GENES_PER_CHUNK; ++g) {
        const float* ap = anchor + ((size_t)arow * G_GENES + g) * D_EMB + kbase;
        const float* bp = other  + ((size_t)brow * G_GENES + g) * D_EMB + kbase;
        v2f a0 = *(const v2f*)(ap);
        v2f a1 = *(const v2f*)(ap + 4);
        v2f b0 = *(const v2f*)(bp);
        v2f b1 = *(const v2f*)(bp + 4);

        // squared norms: each lane holds 4 of 8 dims of its row; combine halves.
        // After the xor, lane l holds the full norm of row l%16 -- exactly the
        // lane that must supply the K=8 augmented element (half 0 only).
        float nap = a0.x * a0.x + a0.y * a0.y + a1.x * a1.x + a1.y * a1.y;
        float nbp = b0.x * b0.x + b0.y * b0.y + b1.x * b1.x + b1.y * b1.y;
        float na = nap + __shfl_xor(nap, 16, 32);
        float nb = nbp + __shfl_xor(nbp, 16, 32);

        // fold the -2 of the cross term into the A fragment
        a0 *= -2.0f;
        a1 *= -2.0f;

        // augmented K=8..11 fragments
        v2f a2, b2;
        a2.x = (half == 0) ? na : 0.0f;   // K=8 : ||a||^2   | K=10: 0
        a2.y = one_if_h0;                 // K=9 : 1         | K=11: 0
        b2.x = one_if_h0;                 // K=8 : 1         | K=10: 0
        b2.y = (half == 0) ? nb : 0.0f;   // K=9 : ||e||^2   | K=11: 0

        v8f c = {};
        c = __builtin_amdgcn_wmma_f32_16x16x4_f32(
                false, a0, false, b0, (short)0, c, false, false);
        c = __builtin_amdgcn_wmma_f32_16x16x4_f32(
                false, a1, false, b1, (short)0, c, false, false);
        c = __builtin_amdgcn_wmma_f32_16x16x4_f32(
                false, a2, false, b2, (short)0, c, false, false);

#pragma unroll
        for (int r = 0; r < 8; ++r) {
            // c[r] == dist^2(M, N); clamp tiny negative rounding, fast sqrt
            acc[r] += __builtin_amdgcn_sqrtf(__builtin_fmaxf(c[r], 0.0f));
        }
    }

#pragma unroll
    for (int r = 0; r < 8; ++r) {
        int i = ti * 16 + (half << 3) + r;   // global row
        int j = tj * 16 + lmod;              // global col
        atomicAdd(&out[i * B_NETS + j], acc[r]);
    }
}

// ---------------------------------------------------------------------------
// Kernel 2: row max/min, hinge loss mean, condition flags.
// out[0] = loss ; out[1..64] = condition (0.0/1.0)
// ---------------------------------------------------------------------------
__global__ __launch_bounds__(64) void reduce_kernel(
    const float* __restrict__ dist_ws, float* __restrict__ out) {
    const int i = threadIdx.x;  // 0..63
    const float inv_g = 1.0f / (float)G_GENES;
    const float* pd = dist_ws;
    const float* nd = dist_ws + B_NETS * B_NETS;

    float pmax = -3.402823466e38f;
    float nmin =  3.402823466e38f;
    for (int j = 0; j < B_NETS; ++j) {
        pmax = fmaxf(pmax, pd[i * B_NETS + j]);
        nmin = fminf(nmin, nd[i * B_NETS + j]);
    }
    pmax *= inv_g;
    nmin *= inv_g;

    out[1 + i] = (pmax + MARGIN_F < nmin) ? 1.0f : 0.0f;
    float l = fmaxf(pmax - nmin + MARGIN_F, 0.0f);

    __shared__ float sbuf[64];
    sbuf[i] = l;
    __syncthreads();
    if (i == 0) {
        float s = 0.0f;
        for (int k = 0; k < 64; ++k) s += sbuf[k];
        out[0] = s * (1.0f / 64.0f);
    }
}

// ---------------------------------------------------------------------------
extern "C" void kernel_launch(void* const* d_in, const int* in_sizes, int n_in,
                              void* d_out, int out_size, void* d_ws, size_t ws_size,
                              hipStream_t stream) {
    const float* anchor   = (const float*)d_in[0];
    const float* positive = (const float*)d_in[1];
    const float* negative = (const float*)d_in[2];
    // d_in[3] = batch_ptr (arange * G) -- implied by compile-time B/G, unused.

    float* ws  = (float*)d_ws;
    float* out = (float*)d_out;

    const int nz = 2 * B_NETS * B_NETS;
    zero_ws_kernel<<<(nz + 255) / 256, 256, 0, stream>>>(ws, nz);

    dim3 grid(16, 2, GENE_SPLIT);
    tile_dist_kernel<<<grid, 32, 0, stream>>>(anchor, positive, negative, ws);

    reduce_kernel<<<1, 64, 0, stream>>>(ws, out);
}